// DenseCaps_15333033247136
// MI455X (gfx1250) — compile-verified
//
#include <hip/hip_runtime.h>

#define IN_CAPS 1152
#define IN_DIM 8
#define OUT_CAPS 10
#define OUT_DIM 16
#define M_TOT (OUT_CAPS * OUT_DIM)   // 160
#define BATCH 512
#define EPS_SQ 1e-8f

typedef __attribute__((ext_vector_type(2))) float v2f;
typedef __attribute__((ext_vector_type(8))) float v8f;

// ---------------------------------------------------------------------------
// Kernel 1: u_hat[b,i,m] = sum_k weight[i,m,k] * u[b,i,k]
// One wave per (i, batch-tile-of-16). A = u tile (16x8), loops 10 m-tiles of
// weight (8x16 slices), two chained V_WMMA_F32_16X16X4_F32 per tile (K=8=2*4).
//
// 32-bit A 16x4 layout: lanes 0-15 hold M=lane, VGPR0=K0/VGPR1=K1;
//                       lanes16-31 hold M=lane-16, VGPR0=K2/VGPR1=K3.
// 32-bit B 4x16 layout mirrors it with N=lane%16.
// C/D: VGPR r, lanes 0-15 -> (M=r, N=lane); lanes 16-31 -> (M=r+8, N=lane-16).
// ---------------------------------------------------------------------------
__global__ void __launch_bounds__(256) denscaps_uhat_wmma(
    const float* __restrict__ u,      // [512][1152][8]
    const float* __restrict__ w,      // [1152][160][8]
    float* __restrict__ uhat)         // [512][1152][160]
{
  const int lane = threadIdx.x & 31;
  const int wid  = ((int)blockIdx.x << 3) + ((int)threadIdx.x >> 5); // 8 waves/block
  const int i    = wid >> 5;          // 0..1151
  const int bt   = wid & 31;          // 0..31
  const int b0   = bt << 4;
  const int half = lane >> 4;         // 0/1
  const int ln   = lane & 15;
  const int koff = half << 1;         // K base for this half: 0 or 2

  // A-matrix fragments (u): lane holds u[b0+ln][i][koff..koff+1] and +4
  const float* urow = u + ((size_t)(b0 + ln) * IN_CAPS + i) * IN_DIM;
  v2f a0, a1;
  a0.x = urow[koff];     a0.y = urow[koff + 1];
  a1.x = urow[koff + 4]; a1.y = urow[koff + 5];

  const float* wbase = w + (size_t)i * M_TOT * IN_DIM;
  float* obase = uhat + (size_t)i * M_TOT;
  const size_t bstride = (size_t)IN_CAPS * M_TOT;

  for (int mt = 0; mt < OUT_CAPS; ++mt) {
    const int m0 = mt << 4;
    const float* wr = wbase + (size_t)(m0 + ln) * IN_DIM;
    v2f bm0, bm1;
    bm0.x = wr[koff];     bm0.y = wr[koff + 1];
    bm1.x = wr[koff + 4]; bm1.y = wr[koff + 5];

    v8f c = {0.f, 0.f, 0.f, 0.f, 0.f, 0.f, 0.f, 0.f};
    c = __builtin_amdgcn_wmma_f32_16x16x4_f32(false, a0, false, bm0,
                                              (short)0, c, false, false);
    c = __builtin_amdgcn_wmma_f32_16x16x4_f32(false, a1, false, bm1,
                                              (short)0, c, false, false);
#pragma unroll
    for (int r = 0; r < 8; ++r) {
      const int brow = b0 + r + (half << 3);
      obase[(size_t)brow * bstride + m0 + ln] = c[r];
    }
  }
}

// ---------------------------------------------------------------------------
// Kernel 2: dynamic routing, one workgroup per batch element.
// LDS: b_logits[1152][10], c[1152][10], 3-segment s partials, v, scales.
// All reductions fixed-order -> deterministic.
// ---------------------------------------------------------------------------
__global__ void __launch_bounds__(512) denscaps_route(
    const float* __restrict__ uhat,   // [512][1152][160]
    float* __restrict__ out)          // [512][10][16]
{
  extern __shared__ float smem[];
  float* b_lds  = smem;                           // 1152*10
  float* c_lds  = b_lds + IN_CAPS * OUT_CAPS;     // 1152*10
  float* s_red  = c_lds + IN_CAPS * OUT_CAPS;     // 3*160
  float* v_lds  = s_red + 3 * M_TOT;              // 160
  float* sc_lds = v_lds + M_TOT;                  // 16 (10 used)

  const int t = threadIdx.x;
  const int b = blockIdx.x;
  const float* uh = uhat + (size_t)b * IN_CAPS * M_TOT;

  for (int idx = t; idx < IN_CAPS * OUT_CAPS; idx += 512) b_lds[idx] = 0.f;
  __syncthreads();

  for (int it = 0; it < 3; ++it) {
    // ---- Phase A: (optional) logit update with v from previous iter, then
    //      per-i softmax over the 10 out-caps. Thread handles i, i+512, ...
    for (int i = t; i < IN_CAPS; i += 512) {
      float logit[OUT_CAPS];
      if (it > 0) {
        const float4* row4 = (const float4*)(uh + (size_t)i * M_TOT);
#pragma unroll
        for (int j = 0; j < OUT_CAPS; ++j) {
          float dot = 0.f;
#pragma unroll
          for (int q = 0; q < 4; ++q) {
            const float4 uv = row4[j * 4 + q];
            const float* vv = &v_lds[j * 16 + q * 4];
            dot += uv.x * vv[0] + uv.y * vv[1] + uv.z * vv[2] + uv.w * vv[3];
          }
          const float nb = b_lds[i * OUT_CAPS + j] + dot;
          b_lds[i * OUT_CAPS + j] = nb;
          logit[j] = nb;
        }
      } else {
#pragma unroll
        for (int j = 0; j < OUT_CAPS; ++j) logit[j] = 0.f;
      }
      float mx = logit[0];
#pragma unroll
      for (int j = 1; j < OUT_CAPS; ++j) mx = fmaxf(mx, logit[j]);
      float e[OUT_CAPS];
      float sum = 0.f;
#pragma unroll
      for (int j = 0; j < OUT_CAPS; ++j) { e[j] = __expf(logit[j] - mx); sum += e[j]; }
      const float inv = 1.f / sum;
#pragma unroll
      for (int j = 0; j < OUT_CAPS; ++j) c_lds[i * OUT_CAPS + j] = e[j] * inv;
    }
    __syncthreads();

    // ---- Phase B: s[j,d] = sum_i c[i,j]*u_hat[i, j*16+d]; 3 i-segments x 160
    if (t < 3 * M_TOT) {
      const int seg = t / M_TOT;
      const int jd  = t - seg * M_TOT;
      const int j   = jd >> 4;
      const int i0  = seg * (IN_CAPS / 3);
      float acc = 0.f;
#pragma unroll 4
      for (int i = i0; i < i0 + IN_CAPS / 3; ++i)
        acc += c_lds[i * OUT_CAPS + j] * uh[(size_t)i * M_TOT + jd];
      s_red[t] = acc;
    }
    __syncthreads();

    if (t < M_TOT) s_red[t] = s_red[t] + s_red[M_TOT + t] + s_red[2 * M_TOT + t];
    __syncthreads();

    // ---- squash: scale = |s|^2 / (1+|s|^2) / (|s| + eps)
    if (t < OUT_CAPS) {
      float n2 = 0.f;
#pragma unroll
      for (int d = 0; d < OUT_DIM; ++d) { const float x = s_red[t * OUT_DIM + d]; n2 += x * x; }
      const float nrm = sqrtf(n2);
      sc_lds[t] = n2 / (1.f + n2) / (nrm + EPS_SQ);
    }
    __syncthreads();
    if (t < M_TOT) v_lds[t] = sc_lds[t >> 4] * s_red[t];
    __syncthreads();
  }

  if (t < M_TOT) out[(size_t)b * M_TOT + t] = v_lds[t];
}

extern "C" void kernel_launch(void* const* d_in, const int* in_sizes, int n_in,
                              void* d_out, int out_size, void* d_ws, size_t ws_size,
                              hipStream_t stream) {
  const float* u = (const float*)d_in[0];   // [512,1152,8]
  const float* w = (const float*)d_in[1];   // [1152,160,8]
  float* uhat = (float*)d_ws;               // needs 512*1152*160*4 = 377,487,360 B

  // 1152 i * 32 batch-tiles = 36864 waves; 8 waves (256 thr) per block
  denscaps_uhat_wmma<<<dim3(36864 / 8), dim3(256), 0, stream>>>(u, w, uhat);

  const size_t lds_bytes =
      (size_t)(IN_CAPS * OUT_CAPS * 2 + 3 * M_TOT + M_TOT + 16) * sizeof(float);
  denscaps_route<<<dim3(BATCH), dim3(512), lds_bytes, stream>>>(uhat, (float*)d_out);
}